// FLoss_16518444220592
// MI455X (gfx1250) — compile-verified
//
#include <hip/hip_runtime.h>
#include <hip/hip_bf16.h>

typedef float v2f __attribute__((ext_vector_type(2)));
typedef float v8f __attribute__((ext_vector_type(8)));

#define BLOCK 256
#define GRID  4096

// Wave-level (32-lane) sum reduction using V_WMMA_F32_16X16X4_F32.
// A (16x4): lane L (L<16) holds row M=L, K=0..1 in VGPR0..1; lane L+16 holds
// the SAME row M=L, K=2..3.  Putting acc in VGPR0 and 0 in VGPR1 gives
// row M = {acc[M], 0, acc[M+16], 0}.  With B = all-ones (layout-agnostic),
// D[m][n] = acc[m] + acc[m+16] for every n.  D VGPR r: lanes 0-15 hold M=r,
// lanes 16-31 hold M=r+8, so sum of a lane's 8 D VGPRs is a half-wave sum;
// one shfl_xor(16) finishes the 32-lane total (in every lane).
// EXEC must be all 1s: call unconditionally from every thread.
__device__ __forceinline__ float wave_sum_wmma(float v) {
    v2f a; a.x = v;    a.y = 0.0f;
    v2f b; b.x = 1.0f; b.y = 1.0f;
    v8f c = {};
    v8f d = __builtin_amdgcn_wmma_f32_16x16x4_f32(
        /*neg_a=*/false, a, /*neg_b=*/false, b,
        /*c_mod=*/(short)0, c, /*reuse_a=*/false, /*reuse_b=*/false);
    float s = d[0] + d[1] + d[2] + d[3] + d[4] + d[5] + d[6] + d[7];
    s += __shfl_xor(s, 16, 32);   // combine the two 8-row halves
    return s;                     // total in all lanes
}

__device__ __forceinline__ float block_sum(float acc) {
    __shared__ float swave[BLOCK / 32];
    const int lane = threadIdx.x & 31;
    const int wid  = threadIdx.x >> 5;
    float ws = wave_sum_wmma(acc);          // EXEC all-1s here
    if (lane == 0) swave[wid] = ws;
    __syncthreads();
    float t = 0.0f;
    if (threadIdx.x == 0) {
        #pragma unroll
        for (int i = 0; i < BLOCK / 32; ++i) t += swave[i];
    }
    return t;                                // valid in thread 0 only
}

// Stage 1: stream output[N,2] (float4 = 2 rows) + target[N] (int4 = 2 int64),
// accumulate (1 - x_t)^2 per thread, write one f32 partial per block.
__global__ void floss_partial_kernel(const float* __restrict__ outp,
                                     const long long* __restrict__ tgt,
                                     float* __restrict__ partials,
                                     long long n) {
    const float4* __restrict__ o4 = reinterpret_cast<const float4*>(outp);
    const int4*   __restrict__ t4 = reinterpret_cast<const int4*>(tgt);

    const long long npairs = n >> 1;
    const long long stride = (long long)gridDim.x * blockDim.x;

    float acc = 0.0f;
    for (long long p = (long long)blockIdx.x * blockDim.x + threadIdx.x;
         p < npairs; p += stride) {
        float4 o = o4[p];          // rows 2p and 2p+1 of output
        int4   t = t4[p];          // low/high dwords of target[2p], target[2p+1]
        float x0 = t.x ? o.y : o.x;
        float x1 = t.z ? o.w : o.z;
        float e0 = 1.0f - x0;
        float e1 = 1.0f - x1;
        acc = fmaf(e0, e0, acc);
        acc = fmaf(e1, e1, acc);
    }
    // Odd-N tail (not hit for N = 16M, kept for robustness).
    if ((n & 1) && blockIdx.x == 0 && threadIdx.x == 0) {
        long long j = n - 1;
        float x = tgt[j] ? outp[2 * j + 1] : outp[2 * j];
        float e = 1.0f - x;
        acc = fmaf(e, e, acc);
    }

    float bs = block_sum(acc);
    if (threadIdx.x == 0) partials[blockIdx.x] = bs;
}

// Stage 2: single block reduces GRID partials and writes the mean.
__global__ void floss_final_kernel(const float* __restrict__ partials,
                                   float* __restrict__ out,
                                   int nparts, long long n) {
    float acc = 0.0f;
    for (int i = threadIdx.x; i < nparts; i += blockDim.x)
        acc += partials[i];
    float total = block_sum(acc);
    if (threadIdx.x == 0)
        out[0] = (float)((double)total / (double)n);
}

extern "C" void kernel_launch(void* const* d_in, const int* in_sizes, int n_in,
                              void* d_out, int out_size, void* d_ws, size_t ws_size,
                              hipStream_t stream) {
    const float*     outp = (const float*)d_in[0];      // [N,2] f32
    const long long* tgt  = (const long long*)d_in[1];  // [N] int64
    float*           out  = (float*)d_out;              // scalar f32
    float*           ws   = (float*)d_ws;               // GRID partials

    const long long n = (long long)in_sizes[1];         // N (rows)

    floss_partial_kernel<<<GRID, BLOCK, 0, stream>>>(outp, tgt, ws, n);
    floss_final_kernel<<<1, BLOCK, 0, stream>>>(ws, out, GRID, n);
}